// RotateProject_40072044871984
// MI455X (gfx1250) — compile-verified
//
#include <hip/hip_runtime.h>

typedef float v2f __attribute__((ext_vector_type(2)));
typedef float v8f __attribute__((ext_vector_type(8)));

#define S_LEN 128
#define NBATCH 4

// One wave32 owns 16 consecutive output pixels (fixed b,y; x = xblk..xblk+15).
// z (128 steps) is split into 4 quarters of 32. Lane layout matches the
// 32-bit 16x4 WMMA A-matrix:
//   lanes 0-15  : row M = lane,    quarters k=0 (A.x) and k=1 (A.y)
//   lanes 16-31 : row M = lane-16, quarters k=2 (A.x) and k=3 (A.y)
// One v_wmma_f32_16x16x4_f32 with B = ones reduces the 4 quarters per row;
// D layout puts rows 0-7 in lane 0 (VGPRs 0-7) and rows 8-15 in lane 16.
//
// Batch index lives in blockIdx.y -> volume base is wave-uniform (saddr
// loads). x-neighbor pairs fetched as one b64 (4 gathers/sample); the
// clamp/zero-pad selection is folded into two pair-weights u0/u1 shared by
// all 4 (z,y) rows.
__global__ __launch_bounds__(256) void RotateProject_kernel(
    const float* __restrict__ vol,   // (B,1,S,S,S) f32
    const float* __restrict__ rot,   // (B,3,3) f32
    float* __restrict__ out)         // (B,S,S) f32
{
    const int lane = threadIdx.x & 31;
    const int wave = threadIdx.x >> 5;
    const int b    = blockIdx.y;                 // uniform batch index
    const int tile = blockIdx.x * 8 + wave;      // 1024 tiles per batch
    const int p0b  = tile << 4;                  // first pixel (within batch)
    const int yi_  = p0b >> 7;
    const int xblk = p0b & 127;

    const int row = lane & 15;                   // M / x offset
    const int hl  = lane >> 4;                   // half: z quarters {2*hl, 2*hl+1}

    // ---- per-batch rotation params (uniform -> scalar loads) ----
    const float* Rb = rot + b * 9;
    const float a00 = Rb[0], a01 = Rb[1], a02 = Rb[2];
    const float a10 = Rb[3], a11 = Rb[4], a12 = Rb[5];
    const float a20 = Rb[6], a21 = Rb[7], a22 = Rb[8];

    // inverse via adjugate (f32, matches reference's f32 inv)
    const float c00 = a11 * a22 - a12 * a21;
    const float c01 = -(a10 * a22 - a12 * a20);
    const float c02 = a10 * a21 - a11 * a20;
    const float id  = 1.0f / (a00 * c00 + a01 * c01 + a02 * c02);
    const float i00 = c00 * id;
    const float i01 = (a02 * a21 - a01 * a22) * id;
    const float i02 = (a01 * a12 - a02 * a11) * id;
    const float i10 = c01 * id;
    const float i11 = (a00 * a22 - a02 * a20) * id;
    const float i12 = (a02 * a10 - a00 * a12) * id;
    const float i20 = c02 * id;
    const float i21 = (a01 * a20 - a00 * a21) * id;
    const float i22 = (a00 * a11 - a01 * a10) * id;

    // t = S/2 * (1 - R @ 1);  tinv = -(Rinv @ t)
    const float t0 = 64.0f * (1.0f - (a00 + a01 + a02));
    const float t1 = 64.0f * (1.0f - (a10 + a11 + a12));
    const float t2 = 64.0f * (1.0f - (a20 + a21 + a22));
    const float tix = -(i00 * t0 + i01 * t1 + i02 * t2);
    const float tiy = -(i10 * t0 + i11 * t1 + i12 * t2);
    const float tiz = -(i20 * t0 + i21 * t1 + i22 * t2);

    const float xf = (float)(xblk + row);
    const float yf = (float)yi_;
    const char* __restrict__ volb =
        (const char*)(vol + (size_t)b * (S_LEN * S_LEN * S_LEN)); // uniform

    v2f accA;   // A-matrix operand: two z-quarter partial sums per lane

    #pragma unroll
    for (int s = 0; s < 2; ++s) {
        const float z0 = (float)((hl * 2 + s) * 32);
        // quarter-start coords: src = Rinv @ (x,y,z0) + tinv
        const float bx = fmaf(i00, xf, fmaf(i01, yf, fmaf(i02, z0, tix)));
        const float by = fmaf(i10, xf, fmaf(i11, yf, fmaf(i12, z0, tiy)));
        const float bz = fmaf(i20, xf, fmaf(i21, yf, fmaf(i22, z0, tiz)));
        float acc = 0.0f;

        #pragma unroll 8
        for (int z = 0; z < 32; ++z) {
            const float zf = (float)z;                 // literal under unroll
            const float sx = fmaf(zf, i02, bx);
            const float sy = fmaf(zf, i12, by);
            const float sz = fmaf(zf, i22, bz);

            const float fx = floorf(sx), fy = floorf(sy), fz = floorf(sz);
            const int xi = (int)fx, yi = (int)fy, zi = (int)fz;
            // lerp weights with validity (zero padding) folded in
            float wx1 = sx - fx, wy1 = sy - fy, wz1 = sz - fz;
            float wx0 = 1.0f - wx1, wy0 = 1.0f - wy1, wz0 = 1.0f - wz1;
            wx0 = ((unsigned)xi < (unsigned)S_LEN) ? wx0 : 0.0f;
            wx1 = ((unsigned)(xi + 1) < (unsigned)S_LEN) ? wx1 : 0.0f;
            wy0 = ((unsigned)yi < (unsigned)S_LEN) ? wy0 : 0.0f;
            wy1 = ((unsigned)(yi + 1) < (unsigned)S_LEN) ? wy1 : 0.0f;
            wz0 = ((unsigned)zi < (unsigned)S_LEN) ? wz0 : 0.0f;
            wz1 = ((unsigned)(zi + 1) < (unsigned)S_LEN) ? wz1 : 0.0f;

            // pair base x64 = clamp(xi, 0, 126); fold corner-select into
            // pair weights u0 (-> p.x) and u1 (-> p.y), shared by 4 rows
            const int  x64 = min(max(xi, 0), S_LEN - 2);
            const bool d0  = (xi >= S_LEN - 1);  // xc0 clamped above x64
            const bool d1  = (xi >= 0);          // xc1 == x64 + 1
            const float wsum = wx0 + wx1;
            float u0 = d1 ? wx0 : wsum;
            u0 = d0 ? 0.0f : u0;
            float u1 = d1 ? wx1 : 0.0f;
            u1 = d0 ? wsum : u1;

            const int yc0 = min(max(yi, 0), S_LEN - 1);
            const int yc1 = min(max(yi + 1, 0), S_LEN - 1);
            const int zc0 = min(max(zi, 0), S_LEN - 1);
            const int zc1 = min(max(zi + 1, 0), S_LEN - 1);

            // byte offsets (x4 folded into the shift-add chain)
            const int xb  = x64 << 2;
            const int o00 = (zc0 << 16) + (yc0 << 9) + xb;
            const int o01 = (zc0 << 16) + (yc1 << 9) + xb;
            const int o10 = (zc1 << 16) + (yc0 << 9) + xb;
            const int o11 = (zc1 << 16) + (yc1 << 9) + xb;

            const float2 p00 = *(const float2*)(volb + o00);
            const float2 p01 = *(const float2*)(volb + o01);
            const float2 p10 = *(const float2*)(volb + o10);
            const float2 p11 = *(const float2*)(volb + o11);

            const float s00 = fmaf(p00.y, u1, p00.x * u0);
            const float s01 = fmaf(p01.y, u1, p01.x * u0);
            const float s10 = fmaf(p10.y, u1, p10.x * u0);
            const float s11 = fmaf(p11.y, u1, p11.x * u0);

            const float sy0 = fmaf(s01, wy1, s00 * wy0);
            const float sy1 = fmaf(s11, wy1, s10 * wy0);
            acc = fmaf(sy1, wz1, fmaf(sy0, wz0, acc));
        }
        accA[s] = acc;
    }

    // Cross-lane reduction of the 4 z-quarters via one WMMA (B = ones).
    // EXEC is all 1s here (uniform control flow, full waves only).
    v2f ones; ones.x = 1.0f; ones.y = 1.0f;
    v8f c = {};
    c = __builtin_amdgcn_wmma_f32_16x16x4_f32(
            /*neg_a=*/false, accA, /*neg_b=*/false, ones,
            /*c_mod=*/(short)0, c, /*reuse_a=*/false, /*reuse_b=*/false);

    // D[m,0]: m=0..7 live in lane 0's VGPRs 0..7, m=8..15 in lane 16's.
    if (row == 0) {
        float* o = out + b * (S_LEN * S_LEN) + p0b + hl * 8;
        ((float4*)o)[0] = ((float4*)&c)[0];
        ((float4*)o)[1] = ((float4*)&c)[1];
    }
}

extern "C" void kernel_launch(void* const* d_in, const int* in_sizes, int n_in,
                              void* d_out, int out_size, void* d_ws, size_t ws_size,
                              hipStream_t stream) {
    const float* vol = (const float*)d_in[0];   // (4,1,128,128,128) f32
    const float* rot = (const float*)d_in[1];   // (4,3,3) f32
    float* out = (float*)d_out;                 // (4,128,128) f32

    // blockIdx.x: 128 tile-groups (8 waves x 16 pixels), blockIdx.y: batch
    RotateProject_kernel<<<dim3(128, NBATCH), 256, 0, stream>>>(vol, rot, out);
}